// HGNNLayer_43559558316711
// MI455X (gfx1250) — compile-verified
//
#include <hip/hip_runtime.h>
#include <hip/hip_bf16.h>

typedef float v2f __attribute__((ext_vector_type(2)));
typedef float v8f __attribute__((ext_vector_type(8)));

#define D 128

// ---------------------------------------------------------------------------
// Phase 1: HX[c] += X[r] * (h_vals * DV_inv_sqrt[r])   (one wave32 per nnz)
// ---------------------------------------------------------------------------
__global__ void __launch_bounds__(256) hgnn_scatter1(
    const float* __restrict__ X, const int* __restrict__ rows,
    const int* __restrict__ cols, const float* __restrict__ vals,
    const float* __restrict__ dvs, float* __restrict__ HX, int nnz) {
  int gid  = blockIdx.x * blockDim.x + threadIdx.x;
  int e    = gid >> 5;          // wave id == nnz index
  int lane = gid & 31;
  if (e >= nnz) return;
  int r = rows[e];
  int c = cols[e];
  float s = vals[e] * dvs[r];
  const float4 x = *(const float4*)(X + (size_t)r * D + lane * 4);
  float* dst = HX + (size_t)c * D + lane * 4;
  unsafeAtomicAdd(dst + 0, x.x * s);
  unsafeAtomicAdd(dst + 1, x.y * s);
  unsafeAtomicAdd(dst + 2, x.z * s);
  unsafeAtomicAdd(dst + 3, x.w * s);
}

// ---------------------------------------------------------------------------
// Phase 2: Xout[r] += HX[c] * (h_vals * DE_inv[c])     (one wave32 per nnz)
// ---------------------------------------------------------------------------
__global__ void __launch_bounds__(256) hgnn_scatter2(
    const float* __restrict__ HX, const int* __restrict__ rows,
    const int* __restrict__ cols, const float* __restrict__ vals,
    const float* __restrict__ des, float* __restrict__ Xout, int nnz) {
  int gid  = blockIdx.x * blockDim.x + threadIdx.x;
  int e    = gid >> 5;
  int lane = gid & 31;
  if (e >= nnz) return;
  int r = rows[e];
  int c = cols[e];
  float s = vals[e] * des[c];
  const float4 h = *(const float4*)(HX + (size_t)c * D + lane * 4);
  float* dst = Xout + (size_t)r * D + lane * 4;
  unsafeAtomicAdd(dst + 0, h.x * s);
  unsafeAtomicAdd(dst + 1, h.y * s);
  unsafeAtomicAdd(dst + 2, h.z * s);
  unsafeAtomicAdd(dst + 3, h.w * s);
}

// ---------------------------------------------------------------------------
// Phase 3: out = (Xout * DV_inv_sqrt[:,None]) @ W^T + b  via WMMA f32 16x16x4
//   Block = 256 threads = 8 waves; each wave computes a 16x128 output stripe.
//   LDS holds W^T packed as K-pairs: Bs[kk*128 + n] = (W[n][2kk], W[n][2kk+1])
//   so one ds_load_b64 yields the B fragment pair per lane.
//   A 16x4 f32 layout: lanes 0-15 -> K=k0,k0+1 ; lanes 16-31 -> K=k0+2,k0+3.
// ---------------------------------------------------------------------------
__global__ void __launch_bounds__(256) hgnn_gemm(
    const float* __restrict__ Xacc, const float* __restrict__ dvs,
    const float* __restrict__ W, const float* __restrict__ b,
    float* __restrict__ out, int Nrows) {
  __shared__ v2f Bs[64 * 128];  // 64 KB: [kk=0..63][n=0..127]

  const int tid = threadIdx.x;
  // Cooperative stage of W^T into LDS (each thread: 32 contiguous-pair loads)
  for (int i = tid; i < 64 * 128; i += 256) {
    int kk = i >> 7;
    int n  = i & 127;
    Bs[i] = *(const v2f*)(W + (size_t)n * D + kk * 2);  // (W[n][2kk], W[n][2kk+1])
  }
  __syncthreads();

  const int wave = tid >> 5;
  const int lane = tid & 31;
  const int row0 = blockIdx.x * 128 + wave * 16;
  if (row0 >= Nrows) return;  // wave-uniform exit; EXEC stays all-1s for WMMA

  const int m  = lane & 15;   // row within tile (A) / col within tile (B,C,D)
  const int kh = lane >> 4;   // K-half selector

  const int   r  = row0 + m;
  const float dv = dvs[r];

  v8f acc[8];
#pragma unroll
  for (int t = 0; t < 8; ++t) acc[t] = (v8f){0.f, 0.f, 0.f, 0.f, 0.f, 0.f, 0.f, 0.f};

  const v2f* arow = (const v2f*)(Xacc + (size_t)r * D);

#pragma unroll 4
  for (int k0 = 0; k0 < D; k0 += 4) {
    const int kkb = (k0 >> 1) + kh;     // pair index this lane consumes
    v2f av = arow[kkb];                 // (A[r][k0+2kh], A[r][k0+2kh+1])
    av *= dv;                           // fuse DV_inv_sqrt scaling
#pragma unroll
    for (int t = 0; t < 8; ++t) {
      v2f bv = Bs[kkb * 128 + t * 16 + m];
      acc[t] = __builtin_amdgcn_wmma_f32_16x16x4_f32(
          /*neg_a=*/false, av, /*neg_b=*/false, bv,
          /*c_mod=*/(short)0, acc[t], /*reuse_a=*/false, /*reuse_b=*/false);
    }
  }

  // D layout: VGPR v -> row (kh*8 + v), col (m) of the 16x16 tile
#pragma unroll
  for (int t = 0; t < 8; ++t) {
    const int nidx = t * 16 + m;
    const float bias = b[nidx];
    float* o = out + (size_t)(row0 + kh * 8) * D + nidx;
#pragma unroll
    for (int v = 0; v < 8; ++v) o[(size_t)v * D] = acc[t][v] + bias;
  }
}

// ---------------------------------------------------------------------------
extern "C" void kernel_launch(void* const* d_in, const int* in_sizes, int n_in,
                              void* d_out, int out_size, void* d_ws, size_t ws_size,
                              hipStream_t stream) {
  const float* X    = (const float*)d_in[0];
  const int*   rows = (const int*)  d_in[1];
  const int*   cols = (const int*)  d_in[2];
  const float* vals = (const float*)d_in[3];
  const float* dvs  = (const float*)d_in[4];
  const float* des  = (const float*)d_in[5];
  const float* W    = (const float*)d_in[6];
  const float* b    = (const float*)d_in[7];
  float* out = (float*)d_out;

  const int nnz = in_sizes[1];
  const int N   = in_sizes[4];
  const int E   = in_sizes[5];

  float* HX   = (float*)d_ws;                 // [E, D]
  float* Xacc = HX + (size_t)E * D;           // [N, D]

  hipMemsetAsync(d_ws, 0, (size_t)(E + N) * D * sizeof(float), stream);

  const int waves_blocks = (nnz * 32 + 255) / 256;
  hgnn_scatter1<<<waves_blocks, 256, 0, stream>>>(X, rows, cols, vals, dvs, HX, nnz);
  hgnn_scatter2<<<waves_blocks, 256, 0, stream>>>(HX, rows, cols, vals, des, Xacc, nnz);

  const int gemm_blocks = (N + 127) / 128;    // 128 rows (8 waves x 16) per block
  hgnn_gemm<<<gemm_blocks, 256, 0, stream>>>(Xacc, dvs, W, b, out, N);
}